// MultiScaleMovingAvg_15745350107490
// MI455X (gfx1250) — compile-verified
//
#include <hip/hip_runtime.h>
#include <hip/hip_bf16.h>

// MultiScaleMovingAvg on MI455X (gfx1250), formulated as a banded-matrix
// matmul so the whole filter runs on the WMMA pipe:
//   Out[s,f] = sum_j g[j-s+15] * x_edgepad[j, f]
// Per 16x16 output tile: 12 x V_WMMA_F32_16X16X4_F32 over a 48-wide K window.
// Edge padding is folded into clamped LDS staging, so the A (filter) matrix
// fragments are constant per lane and computed once.

typedef float v2f __attribute__((ext_vector_type(2)));
typedef float v8f __attribute__((ext_vector_type(8)));

#define B_DIM 16
#define S_DIM 4096
#define F_DIM 512

#define TILE_S 128            // S extent per block (8 M-tiles of 16)
#define TILE_F 32             // F extent per block (2 N-tiles of 16)
#define HALO   15             // (31-1)/2
#define NROWS  160            // 128 + 30 halo, rounded up to K-step padding
#define LDS_STRIDE 40         // floats; 2*40 % 64 == 16 -> conflict-free B reads
#define NTHREADS 256          // 8 waves (wave32)

__global__ __launch_bounds__(NTHREADS)
void msma_wmma_kernel(const float* __restrict__ x,
                      const float* __restrict__ kw,
                      float* __restrict__ out) {
    __shared__ float lds[NROWS * LDS_STRIDE];   // 25.6 KB

    const int tid  = threadIdx.x;
    const int lane = tid & 31;
    const int wave = tid >> 5;
    const int lo   = lane & 15;     // n (B,C,D) / m (A) index within lane group
    const int hi   = lane >> 4;     // lane-half select

    const int sBase = blockIdx.x * TILE_S;
    const int f0    = blockIdx.y * TILE_F;
    const int b     = blockIdx.z;

    // ---- softmax over the 4 scale weights -> per-scale box coefficients ----
    const float k0 = kw[0], k1 = kw[1], k2 = kw[2], k3 = kw[3];
    const float mx = fmaxf(fmaxf(k0, k1), fmaxf(k2, k3));
    const float e0 = __expf(k0 - mx), e1 = __expf(k1 - mx);
    const float e2 = __expf(k2 - mx), e3 = __expf(k3 - mx);
    const float inv = 1.0f / (e0 + e1 + e2 + e3);
    const float c0 = e0 * inv * (1.0f / 3.0f);
    const float c1 = e1 * inv * (1.0f / 7.0f);
    const float c2 = e2 * inv * (1.0f / 15.0f);
    const float c3 = e3 * inv * (1.0f / 31.0f);

    // ---- constant A (banded filter) fragments, WMMA 16x4 f32 layout:
    // lane L, vgpr v holds A[m = L%16][k = v + 2*(L/16)] of each 4-wide K step.
    // A[m][jj] = g(jj - m), g(t) = filter weight at offset t-15 (0 outside band).
    v2f afrag[12];
#pragma unroll
    for (int kk = 0; kk < 12; ++kk) {
#pragma unroll
        for (int v = 0; v < 2; ++v) {
            int t = kk * 4 + v + 2 * hi - lo;   // jj - m
            int d = t - 15; d = (d < 0) ? -d : d;
            float g = 0.0f;
            g += (d <= 1)  ? c0 : 0.0f;
            g += (d <= 3)  ? c1 : 0.0f;
            g += (d <= 7)  ? c2 : 0.0f;
            g += (d <= 15) ? c3 : 0.0f;
            afrag[kk][v] = g;
        }
    }

    // ---- stage rows [sBase-15, sBase+TILE_S+17) x 32 channels into LDS ----
    // Edge padding via clamped row index (this is what keeps A tile-invariant).
    for (int idx = tid; idx < NROWS * (TILE_F / 4); idx += NTHREADS) {
        int row = idx >> 3;                 // 8 float4 per 32-ch row
        int c4  = idx & 7;
        int s = sBase - HALO + row;
        s = (s < 0) ? 0 : ((s > S_DIM - 1) ? (S_DIM - 1) : s);
        const float4 v4 = *reinterpret_cast<const float4*>(
            x + ((size_t)b * S_DIM + s) * F_DIM + f0 + c4 * 4);
        *reinterpret_cast<float4*>(&lds[row * LDS_STRIDE + c4 * 4]) = v4;
    }
    __syncthreads();

    // ---- 16 tiles (8 M x 2 N) across 8 waves, 2 tiles per wave ----
#pragma unroll
    for (int rep = 0; rep < 2; ++rep) {
        const int tileId = wave + rep * 8;
        const int mtile  = tileId >> 1;
        const int ntile  = tileId & 1;
        const int ms     = mtile * 16;            // K window = LDS rows [ms, ms+47]
        const int ncol   = ntile * 16 + lo;

        v8f acc = {0.f, 0.f, 0.f, 0.f, 0.f, 0.f, 0.f, 0.f};
#pragma unroll
        for (int kk = 0; kk < 12; ++kk) {
            const int j0 = ms + kk * 4 + 2 * hi;  // B 4x16 f32 layout: k = v + 2*hi
            v2f bfrag;
            bfrag[0] = lds[j0 * LDS_STRIDE + ncol];
            bfrag[1] = lds[(j0 + 1) * LDS_STRIDE + ncol];
            acc = __builtin_amdgcn_wmma_f32_16x16x4_f32(
                false, afrag[kk], false, bfrag, (short)0, acc, false, false);
        }

        // ---- store D: lane L, vgpr v -> (m = v + 8*hi, n = lo) ----
        const int sOut = sBase + ms + 8 * hi;
        const int fOut = f0 + ntile * 16 + lo;
        float* o = out + ((size_t)b * S_DIM + sOut) * F_DIM + fOut;
#pragma unroll
        for (int v = 0; v < 8; ++v) {
            o[(size_t)v * F_DIM] = acc[v];
        }
    }
}

extern "C" void kernel_launch(void* const* d_in, const int* in_sizes, int n_in,
                              void* d_out, int out_size, void* d_ws, size_t ws_size,
                              hipStream_t stream) {
    const float* x  = (const float*)d_in[0];
    const float* kw = (const float*)d_in[1];
    float* out = (float*)d_out;

    dim3 grid(S_DIM / TILE_S, F_DIM / TILE_F, B_DIM);  // (32, 16, 16)
    msma_wmma_kernel<<<grid, NTHREADS, 0, stream>>>(x, kw, out);
}